// SimplifiedNormalEstimator_5274219840017
// MI455X (gfx1250) — compile-verified
//
#include <hip/hip_runtime.h>
#include <math.h>

typedef __attribute__((ext_vector_type(2))) float v2f;
typedef __attribute__((ext_vector_type(8))) float v8f;

#define NB     4
#define NPTS   8192
#define KNN    30
#define CHUNK  128          // columns of distance tile staged in LDS per wave
#define WAVES  8
#define RPW    16           // query rows per wave (one WMMA M-tile)

// Register-resident top-K (smallest) set, unsorted, with tracked max.
// Fully unrolled so bd/bi stay in VGPRs (no dynamic register indexing).
__device__ __forceinline__ void topk_insert(float (&bd)[KNN], int (&bi)[KNN],
                                            float &cmax, int &cpos,
                                            float d, int idx) {
  if (d < cmax) {
#pragma unroll
    for (int k = 0; k < KNN; k++)
      if (k == cpos) { bd[k] = d; bi[k] = idx; }
    float m = bd[0]; int p = 0;
#pragma unroll
    for (int k = 1; k < KNN; k++)
      if (bd[k] > m) { m = bd[k]; p = k; }
    cmax = m; cpos = p;
  }
}

__global__ __launch_bounds__(256)
void knn_normals_kernel(const float* __restrict__ pts_all,
                        float* __restrict__ out) {
  // Per-wave region: 16 rows x CHUNK cols of f32 distances (2048 f32 = 8KB).
  // Merge buffers (16x32 f32 + 16x32 i32 = 4KB) are overlaid on the same region
  // after the chunk loop finishes.
  __shared__ float smem[WAVES * RPW * CHUNK];   // 64 KB

  const int tid  = threadIdx.x;
  const int wave = tid >> 5;
  const int lane = tid & 31;
  const int lh   = lane >> 4;   // lane half: selects K-pair (A/B frag) / row+8 (D frag)
  const int ls   = lane & 15;

  const int blocksPerBatch = NPTS / (WAVES * RPW);       // 64
  const int b     = blockIdx.x / blocksPerBatch;
  const int qtile = blockIdx.x % blocksPerBatch;
  const int qbase = qtile * (WAVES * RPW) + wave * RPW;  // batch-local
  const float* __restrict__ pts = pts_all + (size_t)b * NPTS * 3;

  // ---- A fragment (16x4 f32, M=queries, K=coords+pad). Loop invariant. ----
  // lane holds A[ls][2*lh + {0,1}] : low half (x,y), high half (z,0)
  const int q = qbase + ls;
  const float qx = pts[q * 3 + 0];
  const float qy = pts[q * 3 + 1];
  const float qz = pts[q * 3 + 2];
  v2f afrag;
  afrag.x = lh ? qz : qx;
  afrag.y = lh ? 0.0f : qy;
  const float qsq_own = qx * qx + qy * qy + qz * qz;

  // |q|^2 per D-fragment VGPR slot v (row = v + 8*lh), fetched cross-lane.
  float qsq_sel[8];
#pragma unroll
  for (int v = 0; v < 8; v++)
    qsq_sel[v] = __shfl(qsq_own, v + 8 * lh, 32);

  // ---- per-lane running top-30 ----
  float bd[KNN]; int bi[KNN];
#pragma unroll
  for (int k = 0; k < KNN; k++) { bd[k] = 3.402823466e38f; bi[k] = 0; }
  float cmax = 3.402823466e38f; int cpos = 0;

  float* wdist = &smem[wave * RPW * CHUNK];
  const int myrow    = ls;                 // row this lane runs selection for
  const int halfbase = lh * (CHUNK / 2);   // lane pair splits columns

  for (int cb = 0; cb < NPTS; cb += CHUNK) {
    __syncthreads();   // protect LDS region from previous iteration's reads

    // ---- WMMA phase: 8 tiles of 16x16 distances into LDS ----
#pragma unroll
    for (int t = 0; t < CHUNK / 16; t++) {
      const int c  = cb + t * 16 + ls;     // column point
      const float cx = pts[c * 3 + 0];
      const float cy = pts[c * 3 + 1];
      const float cz = pts[c * 3 + 2];
      v2f bfrag;                           // B[2*lh + {0,1}][ls]
      bfrag.x = lh ? cz : cx;
      bfrag.y = lh ? 0.0f : cy;
      const float csq = cx * cx + cy * cy + cz * cz;

      v8f acc = {};
      acc = __builtin_amdgcn_wmma_f32_16x16x4_f32(
          /*neg_a=*/false, afrag, /*neg_b=*/false, bfrag,
          /*c_mod=*/(short)0, acc, /*reuse_a=*/false, /*reuse_b=*/false);

      // d2[row][col] = |q|^2 + |c|^2 - 2*dot ; D frag: row = v + 8*lh, col = ls
#pragma unroll
      for (int v = 0; v < 8; v++) {
        const int row = v + 8 * lh;
        const float d2 = qsq_sel[v] + csq - 2.0f * acc[v];
        wdist[row * CHUNK + t * 16 + ls] = d2;
      }
    }
    __syncthreads();

    // ---- selection phase: lane scans its 64-column half of its row ----
    const float4* rp =
        reinterpret_cast<const float4*>(&wdist[myrow * CHUNK + halfbase]);
    for (int j = 0; j < (CHUNK / 2) / 4; j++) {
      const float4 dv = rp[j];
      const int c0 = cb + halfbase + j * 4;
      if (c0 + 0 != q) topk_insert(bd, bi, cmax, cpos, dv.x, c0 + 0);
      if (c0 + 1 != q) topk_insert(bd, bi, cmax, cpos, dv.y, c0 + 1);
      if (c0 + 2 != q) topk_insert(bd, bi, cmax, cpos, dv.z, c0 + 2);
      if (c0 + 3 != q) topk_insert(bd, bi, cmax, cpos, dv.w, c0 + 3);
    }
  }

  // ---- merge lane pair lists: upper half publishes, lower half absorbs ----
  __syncthreads();
  float* mD = &smem[wave * RPW * CHUNK];          // 16*32 floats
  int*   mI = reinterpret_cast<int*>(mD + RPW * 32);
  if (lh == 1) {
#pragma unroll
    for (int k = 0; k < KNN; k++) { mD[ls * 32 + k] = bd[k]; mI[ls * 32 + k] = bi[k]; }
  }
  __syncthreads();

  if (lh == 0) {
#pragma unroll
    for (int k = 0; k < KNN; k++)
      topk_insert(bd, bi, cmax, cpos, mD[ls * 32 + k], mI[ls * 32 + k]);

    // ---- covariance of the 30 nearest neighbors (order-invariant) ----
    float sx = 0, sy = 0, sz = 0;
    float sxx = 0, sxy = 0, sxz = 0, syy = 0, syz = 0, szz = 0;
#pragma unroll
    for (int k = 0; k < KNN; k++) {
      const int j = bi[k];
      const float x = pts[j * 3 + 0];
      const float y = pts[j * 3 + 1];
      const float z = pts[j * 3 + 2];
      sx += x; sy += y; sz += z;
      sxx += x * x; sxy += x * y; sxz += x * z;
      syy += y * y; syz += y * z; szz += z * z;
    }
    const float inv = 1.0f / (float)KNN;
    const float mx = sx * inv, my = sy * inv, mz = sz * inv;
    const float cxx = sxx * inv - mx * mx;
    const float cyy = syy * inv - my * my;
    const float czz = szz * inv - mz * mz;
    const float cxy = sxy * inv - mx * my;
    const float cxz = sxz * inv - mx * mz;
    const float cyz = syz * inv - my * mz;

    // ---- smallest eigenvalue of symmetric 3x3 (Smith's trig method) ----
    const float p1 = cxy * cxy + cxz * cxz + cyz * cyz;
    const float qm = (cxx + cyy + czz) * (1.0f / 3.0f);
    const float dxx = cxx - qm, dyy = cyy - qm, dzz = czz - qm;
    const float p2 = dxx * dxx + dyy * dyy + dzz * dzz + 2.0f * p1;
    const float p  = sqrtf(p2 * (1.0f / 6.0f));
    float lam = qm;
    if (p > 1e-20f) {
      const float ip = 1.0f / p;
      const float bxx = dxx * ip, byy = dyy * ip, bzz = dzz * ip;
      const float bxy = cxy * ip, bxz = cxz * ip, byz = cyz * ip;
      float r = 0.5f * (bxx * (byy * bzz - byz * byz)
                      - bxy * (bxy * bzz - byz * bxz)
                      + bxz * (bxy * byz - byy * bxz));
      r = fminf(1.0f, fmaxf(-1.0f, r));
      const float phi = acosf(r) * (1.0f / 3.0f);
      lam = qm + 2.0f * p * cosf(phi + 2.09439510239319549f); // +2*pi/3 -> min
    }

    // ---- eigenvector: largest cross product of rows of (C - lam*I) ----
    const float m00 = cxx - lam, m11 = cyy - lam, m22 = czz - lam;
    const float v0x = cxy * cyz - cxz * m11;
    const float v0y = cxz * cxy - m00 * cyz;
    const float v0z = m00 * m11 - cxy * cxy;
    const float v1x = cxy * m22 - cxz * cyz;
    const float v1y = cxz * cxz - m00 * m22;
    const float v1z = m00 * cyz - cxy * cxz;
    const float v2x = m11 * m22 - cyz * cyz;
    const float v2y = cyz * cxz - cxy * m22;
    const float v2z = cxy * cyz - m11 * cxz;
    const float n0 = v0x * v0x + v0y * v0y + v0z * v0z;
    const float n1 = v1x * v1x + v1y * v1y + v1z * v1z;
    const float n2 = v2x * v2x + v2y * v2y + v2z * v2z;
    float nx, ny, nz, nn;
    if (n0 >= n1 && n0 >= n2)      { nx = v0x; ny = v0y; nz = v0z; nn = n0; }
    else if (n1 >= n2)             { nx = v1x; ny = v1y; nz = v1z; nn = n1; }
    else                           { nx = v2x; ny = v2y; nz = v2z; nn = n2; }
    if (nn < 1e-30f) { nx = 0.0f; ny = 0.0f; nz = 1.0f; nn = 1.0f; }
    const float rcp = 1.0f / fmaxf(sqrtf(nn), 1e-12f);

    float* o = out + (size_t)(b * NPTS + q) * 3;
    o[0] = nx * rcp;
    o[1] = ny * rcp;
    o[2] = nz * rcp;
  }
}

extern "C" void kernel_launch(void* const* d_in, const int* in_sizes, int n_in,
                              void* d_out, int out_size, void* d_ws, size_t ws_size,
                              hipStream_t stream) {
  (void)in_sizes; (void)n_in; (void)out_size; (void)d_ws; (void)ws_size;
  const float* pts = (const float*)d_in[0];
  float* out = (float*)d_out;
  const int blocks = NB * (NPTS / (WAVES * RPW));   // 4 * 64 = 256 blocks
  knn_normals_kernel<<<blocks, 256, 0, stream>>>(pts, out);
}